// GNN_68367289417838
// MI455X (gfx1250) — compile-verified
//
#include <hip/hip_runtime.h>
#include <hip/hip_bf16.h>
#include <math.h>

// ---------------- constants (match reference) ----------------
#define HD   128
#define TYP  3
#define REL  4
#define NHD  8
#define DKD  16
#define NLAY 2
#define MAXT 240

typedef float v2f __attribute__((ext_vector_type(2)));
typedef float v8f __attribute__((ext_vector_type(8)));
typedef unsigned int v4u __attribute__((ext_vector_type(4)));
typedef int v4i __attribute__((ext_vector_type(4)));
typedef int v8i __attribute__((ext_vector_type(8)));

// monotone float<->uint encoding for atomicMax on floats
__device__ __forceinline__ unsigned encF(float f) {
  unsigned u = __float_as_uint(f);
  return (u & 0x80000000u) ? ~u : (u | 0x80000000u);
}
__device__ __forceinline__ float decF(unsigned u) {
  return (u & 0x80000000u) ? __uint_as_float(u & 0x7FFFFFFFu) : __uint_as_float(~u);
}
#define ENC_NEG_INF 0x007FFFFFu  // encF(-inf)

// =====================================================================
// TDM: DMA a [tileRows x K] f32 tile (row stride K elements) from global
// into LDS at byte offset ldsOff, padding each row by 4 DWORDs so the
// LDS row stride is K+4 (bank-conflict-free for 16-row column reads).
// tensor_dim1 = rowsAvail => rows past the end are OOB and return zero.
// D# bitfields per CDNA5 ISA ch.8 (group0/group1), 2-D tensor: groups
// 2/3 zero. amdgpu-toolchain (clang-23) 6-arg builtin form.
// =====================================================================
__device__ __forceinline__ void tdm_load_tile(const float* gsrc, unsigned ldsOff,
                                              int K, int rowsAvail, int tileRows) {
  unsigned long long ga = (unsigned long long)(size_t)gsrc;
  v4u g0;
  g0[0] = 1u;                                                  // count=1 (valid), user mode
  g0[1] = ldsOff;                                              // lds_addr (bytes)
  g0[2] = (unsigned)(ga & 0xFFFFFFFFu);                        // global_addr[31:0]
  g0[3] = (unsigned)((ga >> 32) & 0x01FFFFFFu) | (2u << 30);   // global_addr[56:32] | type=2
  const int interval = (K == 256) ? 7 : 6;  // pad every 256 / 128 DWORDs (= one row)
  v8i g1;
  g1[0] = (2 << 16)            // data_size = 4 bytes
        | (1 << 20)            // pad_enable
        | (interval << 22)     // pad_interval
        | (3 << 25);           // pad_amount = 4 DWORDs
  g1[1] = (K & 0xFFFF) << 16;                                  // tensor_dim0[15:0]
  g1[2] = ((unsigned)K >> 16) | ((rowsAvail & 0xFFFF) << 16);  // tensor_dim0 hi | tensor_dim1 lo
  g1[3] = ((unsigned)rowsAvail >> 16) | ((K & 0xFFFF) << 16);  // tensor_dim1 hi | tile_dim0
  g1[4] = (tileRows & 0xFFFF);                                 // tile_dim1 (tile_dim2 = 0)
  g1[5] = K;                                                   // tensor_dim0_stride[31:0]
  g1[6] = 0;                                                   // stride hi / dim1_stride lo
  g1[7] = 0;
  v4i z4 = {0, 0, 0, 0};
  v8i z8 = {0, 0, 0, 0, 0, 0, 0, 0};
  __builtin_amdgcn_tensor_load_to_lds(g0, g1, z4, z4, z8, 0);
}

// =====================================================================
// Generic per-type linear: Y[M,128] = act(X[M,K] @ W[z][K,128] + b[z])
// gridDim.z = T (types != null, masked write) or 1 (types == null).
// One block = 32 rows; 8 waves each own a 16-col tile and TWO 16x16 row
// tiles (2 accumulators share each B fragment). X tile staged by TDM.
// =====================================================================
__global__ void __launch_bounds__(256)
hgt_wmma_linear(const float* __restrict__ X, const float* __restrict__ W,
                const float* __restrict__ bias, const int* __restrict__ types,
                float* __restrict__ Y, int M, int K, int act) {
  __shared__ float Xs[32 * (256 + 4)];
  const int KP = K + 4;
  const int row0 = blockIdx.x * 32;
  const int z = blockIdx.z;
  const float* Wz = W + (size_t)z * K * HD;
  const float* bz = bias ? (bias + (size_t)z * HD) : nullptr;

  // wave 0 DMAs the 32 x K tile into LDS via the Tensor Data Mover
  if (threadIdx.x < 32) {
    tdm_load_tile(X + (size_t)row0 * K, (unsigned)(size_t)Xs, K, M - row0, 32);
    __builtin_amdgcn_s_wait_tensorcnt(0);
  }
  __syncthreads();

  const int w    = threadIdx.x >> 5;   // wave -> 16-col output tile
  const int lane = threadIdx.x & 31;
  const int half = lane >> 4;          // 0: K pair {0,1}; 1: K pair {2,3}
  const int lid  = lane & 15;
  const int col  = w * 16 + lid;

  v8f c0 = {0.f,0.f,0.f,0.f,0.f,0.f,0.f,0.f};
  v8f c1 = {0.f,0.f,0.f,0.f,0.f,0.f,0.f,0.f};
  for (int kk = 0; kk < K; kk += 4) {
    v2f a0, a1, b;
    // A-frags: lane<16 -> K={kk,kk+1}; lane>=16 -> K={kk+2,kk+3} (ISA 16x4 f32 layout)
    const int ao = kk + 2 * half;
    a0[0] = Xs[lid * KP + ao];
    a0[1] = Xs[lid * KP + ao + 1];
    a1[0] = Xs[(lid + 16) * KP + ao];
    a1[1] = Xs[(lid + 16) * KP + ao + 1];
    // B-frag: lane%16 = N column, VGPRs hold the K pair (shared by both row tiles)
    const float* wp = Wz + (size_t)ao * HD + col;
    b[0] = wp[0];
    b[1] = wp[HD];
    __builtin_prefetch(wp + 4 * HD, 0, 0);  // next K-step of W -> global_prefetch_b8
    c0 = __builtin_amdgcn_wmma_f32_16x16x4_f32(false, a0, false, b, (short)0, c0,
                                               false, false);
    c1 = __builtin_amdgcn_wmma_f32_16x16x4_f32(false, a1, false, b, (short)0, c1,
                                               false, false);
  }

  const float bval = bz ? bz[col] : 0.0f;
#pragma unroll
  for (int i = 0; i < 8; ++i) {
    const int rr = i + 8 * half;  // C/D layout: VGPR i -> M=i (+8 for hi half)
    int gr = row0 + rr;
    if (gr < M && (!types || types[gr] == z)) {
      float vv = c0[i] + bval;
      if (act == 1) vv = tanhf(vv);
      Y[(size_t)gr * HD + col] = vv;
    }
    gr = row0 + 16 + rr;
    if (gr < M && (!types || types[gr] == z)) {
      float vv = c1[i] + bval;
      if (act == 1) vv = tanhf(vv);
      Y[(size_t)gr * HD + col] = vv;
    }
  }
}

// =====================================================================
// Per-layer init: aggr=0, z=0, m=-inf(encoded)
// =====================================================================
__global__ void hgt_init_layer(float* aggr, float* zsum, unsigned* mEnc, int N) {
  size_t i = (size_t)blockIdx.x * blockDim.x + threadIdx.x;
  if (i < (size_t)N * HD) aggr[i] = 0.0f;
  if (i < (size_t)N * NHD) { zsum[i] = 0.0f; mEnc[i] = ENC_NEG_INF; }
}

// =====================================================================
// Pass A: per-edge attention score + segment max (wave per edge).
// k_edge = k_nodes[src] + k_rte[st][time]; per-head k' = k @ rel_att[r];
// score = (q[dst]·k') * pri * (1/sqrt(16)).
// =====================================================================
__global__ void __launch_bounds__(256)
hgt_edge_score(const int* __restrict__ esrc, const int* __restrict__ edst,
               const int* __restrict__ etype, const int* __restrict__ etime,
               const int* __restrict__ ntype,
               const float* __restrict__ q_nodes, const float* __restrict__ k_nodes,
               const float* __restrict__ k_rte,     // [T,240,128]
               const float* __restrict__ rel_att_l, // [R,8,16,16]
               const float* __restrict__ rel_pri_l, // [T,R,T,8]
               float* __restrict__ score, unsigned* __restrict__ mEnc, int E) {
  __shared__ float As[REL * NHD * DKD * DKD];  // 32 KB: all 4 relation tables
  __shared__ float kS[8 * HD];                 // per-wave staged k vector
  for (int i = threadIdx.x; i < REL * NHD * DKD * DKD; i += 256)
    As[i] = rel_att_l[i];
  __syncthreads();

  const int w    = threadIdx.x >> 5;
  const int lane = threadIdx.x & 31;
  const long e   = (long)blockIdx.x * 8 + w;
  if (e >= E) return;

  const int s  = esrc[e], d = edst[e], r = etype[e], tm = etime[e];
  const int st = ntype[s], tt = ntype[d];

  float4 kv = *reinterpret_cast<const float4*>(k_nodes + (size_t)s * HD + lane * 4);
  float4 kr = *reinterpret_cast<const float4*>(k_rte + (((size_t)st * MAXT + tm) << 7) + lane * 4);
  kS[w * HD + lane * 4 + 0] = kv.x + kr.x;
  kS[w * HD + lane * 4 + 1] = kv.y + kr.y;
  kS[w * HD + lane * 4 + 2] = kv.z + kr.z;
  kS[w * HD + lane * 4 + 3] = kv.w + kr.w;  // wave-coherent LDS; no barrier needed

  float4 qv = *reinterpret_cast<const float4*>(q_nodes + (size_t)d * HD + lane * 4);
  const float qc[4] = {qv.x, qv.y, qv.z, qv.w};

  const int c0 = lane * 4;
  const int h  = c0 >> 4;           // head (constant over the 4 components)
  const int j0 = c0 & 15;
  const float* kh  = &kS[w * HD + h * DKD];
  const float* Ahb = &As[((r * NHD + h) << 8)];  // 16x16 block

  float part = 0.0f;
#pragma unroll
  for (int jj = 0; jj < 4; ++jj) {
    float acc = 0.0f;
    const float* Acol = Ahb + (j0 + jj);
#pragma unroll
    for (int i = 0; i < DKD; ++i) acc += kh[i] * Acol[i * DKD];
    part += acc * qc[jj];
  }
  // reduce dot over the 4 lanes of this head
  part += __shfl_xor(part, 1, 32);
  part += __shfl_xor(part, 2, 32);

  if ((lane & 3) == 0) {
    const float pri = rel_pri_l[(((size_t)tt * REL + r) * TYP + st) * NHD + h];
    const float sc = part * pri * 0.25f;  // 1/sqrt(DK)
    score[(size_t)e * NHD + h] = sc;
    atomicMax(&mEnc[(size_t)d * NHD + h], encF(sc));
  }
}

// =====================================================================
// Pass B: z[dst,h] += exp(score - m[dst,h])   (thread per edge*head)
// =====================================================================
__global__ void hgt_edge_expsum(const int* __restrict__ edst,
                                const float* __restrict__ score,
                                const unsigned* __restrict__ mEnc,
                                float* __restrict__ zsum, int E) {
  size_t i = (size_t)blockIdx.x * blockDim.x + threadIdx.x;
  if (i >= (size_t)E * NHD) return;
  const int e = (int)(i >> 3), h = (int)(i & 7);
  const int d = edst[e];
  const float mm = decF(mEnc[(size_t)d * NHD + h]);
  unsafeAtomicAdd(&zsum[(size_t)d * NHD + h], __expf(score[i] - mm));
}

// =====================================================================
// Pass C: message + scatter-add (wave per edge).
// v' = (v_nodes[src]+v_rte) @ rel_msg[r]; aggr[dst] += v' * att
// =====================================================================
__global__ void __launch_bounds__(256)
hgt_edge_msg(const int* __restrict__ esrc, const int* __restrict__ edst,
             const int* __restrict__ etype, const int* __restrict__ etime,
             const int* __restrict__ ntype,
             const float* __restrict__ v_nodes, const float* __restrict__ v_rte,
             const float* __restrict__ rel_msg_l,
             const float* __restrict__ score, const unsigned* __restrict__ mEnc,
             const float* __restrict__ zsum, float* __restrict__ aggr, int E) {
  __shared__ float Ms[REL * NHD * DKD * DKD];
  __shared__ float vS[8 * HD];
  for (int i = threadIdx.x; i < REL * NHD * DKD * DKD; i += 256)
    Ms[i] = rel_msg_l[i];
  __syncthreads();

  const int w    = threadIdx.x >> 5;
  const int lane = threadIdx.x & 31;
  const long e   = (long)blockIdx.x * 8 + w;
  if (e >= E) return;

  const int s = esrc[e], d = edst[e], r = etype[e], tm = etime[e];
  const int st = ntype[s];

  float4 vv = *reinterpret_cast<const float4*>(v_nodes + (size_t)s * HD + lane * 4);
  float4 vr = *reinterpret_cast<const float4*>(v_rte + (((size_t)st * MAXT + tm) << 7) + lane * 4);
  vS[w * HD + lane * 4 + 0] = vv.x + vr.x;
  vS[w * HD + lane * 4 + 1] = vv.y + vr.y;
  vS[w * HD + lane * 4 + 2] = vv.z + vr.z;
  vS[w * HD + lane * 4 + 3] = vv.w + vr.w;

  const int c0 = lane * 4;
  const int h  = c0 >> 4;
  const int j0 = c0 & 15;
  const float* vh  = &vS[w * HD + h * DKD];
  const float* Mhb = &Ms[((r * NHD + h) << 8)];

  const float mm  = decF(mEnc[(size_t)d * NHD + h]);
  const float zz  = zsum[(size_t)d * NHD + h];
  const float att = __expf(score[(size_t)e * NHD + h] - mm) / zz;

#pragma unroll
  for (int jj = 0; jj < 4; ++jj) {
    float acc = 0.0f;
    const float* Mcol = Mhb + (j0 + jj);
#pragma unroll
    for (int i = 0; i < DKD; ++i) acc += vh[i] * Mcol[i * DKD];
    unsafeAtomicAdd(&aggr[(size_t)d * HD + c0 + jj], acc * att);
  }
}

// exact GELU (reference uses approximate=False)
__global__ void hgt_gelu(float* __restrict__ a, size_t n) {
  size_t i = (size_t)blockIdx.x * blockDim.x + threadIdx.x;
  if (i >= n) return;
  float x = a[i];
  a[i] = 0.5f * x * (1.0f + erff(x * 0.70710678118654752f));
}

// x = out*alpha + x*(1-alpha), alpha = sigmoid(skip[l][type])
__global__ void hgt_blend(const float* __restrict__ outb, const int* __restrict__ ntype,
                          const float* __restrict__ skip_l, float* __restrict__ x, int N) {
  size_t i = (size_t)blockIdx.x * blockDim.x + threadIdx.x;
  if (i >= (size_t)N * HD) return;
  const int n = (int)(i >> 7);
  const float a = 1.0f / (1.0f + __expf(-skip_l[ntype[n]]));
  x[i] = outb[i] * a + x[i] * (1.0f - a);
}

// =====================================================================
extern "C" void kernel_launch(void* const* d_in, const int* in_sizes, int n_in,
                              void* d_out, int out_size, void* d_ws, size_t ws_size,
                              hipStream_t stream) {
  const float* node_feature = (const float*)d_in[0];
  const int*   node_type    = (const int*)d_in[1];
  const int*   edge_time    = (const int*)d_in[2];
  const int*   edge_type    = (const int*)d_in[3];
  const int*   edge_index   = (const int*)d_in[4];
  const float* adapt_W = (const float*)d_in[5];
  const float* adapt_b = (const float*)d_in[6];
  const float* Wk = (const float*)d_in[7];  const float* bk = (const float*)d_in[8];
  const float* Wq = (const float*)d_in[9];  const float* bq = (const float*)d_in[10];
  const float* Wv = (const float*)d_in[11]; const float* bv = (const float*)d_in[12];
  const float* Wa = (const float*)d_in[13]; const float* ba = (const float*)d_in[14];
  const float* rel_pri = (const float*)d_in[15];
  const float* rel_att = (const float*)d_in[16];
  const float* rel_msg = (const float*)d_in[17];
  const float* skip    = (const float*)d_in[18];
  const float* rte_emb = (const float*)d_in[19];
  const float* rte_W   = (const float*)d_in[20];
  const float* rte_b   = (const float*)d_in[21];

  const int N  = in_sizes[1];
  const int E  = in_sizes[2];
  const int IN = in_sizes[0] / N;
  const int* esrc = edge_index;
  const int* edst = edge_index + E;

  float* x = (float*)d_out;  // current node features live in d_out

  // workspace layout (floats)
  float* q       = (float*)d_ws;
  float* k       = q       + (size_t)N * HD;
  float* v       = k       + (size_t)N * HD;
  float* aggr    = v       + (size_t)N * HD;
  float* outb    = aggr    + (size_t)N * HD;
  float* rte_tab = outb    + (size_t)N * HD;           // [240,128]
  float* k_rte   = rte_tab + (size_t)MAXT * HD;        // [3,240,128]
  float* v_rte   = k_rte   + (size_t)TYP * MAXT * HD;
  float* score   = v_rte   + (size_t)TYP * MAXT * HD;  // [E,8]
  float* zsum    = score   + (size_t)E * NHD;          // [N,8]
  unsigned* mEnc = (unsigned*)(zsum + (size_t)N * NHD);

  const dim3 blk(256);
  const dim3 gN32T((N + 31) / 32, 1, TYP);
  const dim3 gRte((MAXT + 31) / 32, 1, 1);
  const dim3 gEdge((unsigned)((E + 7) / 8));
  const unsigned gNH = (unsigned)(((size_t)N * HD + 255) / 256);
  const unsigned gEH = (unsigned)(((size_t)E * NHD + 255) / 256);

  // adapt: x = tanh(per_type_linear(node_feature))
  hipLaunchKernelGGL(hgt_wmma_linear, gN32T, blk, 0, stream,
                     node_feature, adapt_W, adapt_b, node_type, x, N, IN, 1);

  for (int l = 0; l < NLAY; ++l) {
    const float* Wk_l = Wk + (size_t)l * TYP * HD * HD;
    const float* Wq_l = Wq + (size_t)l * TYP * HD * HD;
    const float* Wv_l = Wv + (size_t)l * TYP * HD * HD;
    const float* Wa_l = Wa + (size_t)l * TYP * HD * HD;
    const float* bk_l = bk + (size_t)l * TYP * HD;
    const float* bq_l = bq + (size_t)l * TYP * HD;
    const float* bv_l = bv + (size_t)l * TYP * HD;
    const float* ba_l = ba + (size_t)l * TYP * HD;
    const float* ratt_l = rel_att + (size_t)l * REL * NHD * DKD * DKD;
    const float* rmsg_l = rel_msg + (size_t)l * REL * NHD * DKD * DKD;
    const float* rpri_l = rel_pri + (size_t)l * TYP * REL * TYP * NHD;

    // rte_tab = rte_emb @ rte_W[l] + rte_b[l]   (240 x 256 x 128)
    hipLaunchKernelGGL(hgt_wmma_linear, gRte, blk, 0, stream,
                       rte_emb, rte_W + (size_t)l * 2 * HD * HD,
                       rte_b + (size_t)l * HD, (const int*)nullptr,
                       rte_tab, MAXT, 2 * HD, 0);
    // k_rte[t] = rte_tab @ Wk[l][t] ; v_rte[t] = rte_tab @ Wv[l][t] (bias lives in node terms)
    for (int t = 0; t < TYP; ++t) {
      hipLaunchKernelGGL(hgt_wmma_linear, gRte, blk, 0, stream,
                         rte_tab, Wk_l + (size_t)t * HD * HD, (const float*)nullptr,
                         (const int*)nullptr, k_rte + (size_t)t * MAXT * HD, MAXT, HD, 0);
      hipLaunchKernelGGL(hgt_wmma_linear, gRte, blk, 0, stream,
                         rte_tab, Wv_l + (size_t)t * HD * HD, (const float*)nullptr,
                         (const int*)nullptr, v_rte + (size_t)t * MAXT * HD, MAXT, HD, 0);
    }

    // node-level q/k/v (per-type GEMMs)
    hipLaunchKernelGGL(hgt_wmma_linear, gN32T, blk, 0, stream, x, Wq_l, bq_l, node_type, q, N, HD, 0);
    hipLaunchKernelGGL(hgt_wmma_linear, gN32T, blk, 0, stream, x, Wk_l, bk_l, node_type, k, N, HD, 0);
    hipLaunchKernelGGL(hgt_wmma_linear, gN32T, blk, 0, stream, x, Wv_l, bv_l, node_type, v, N, HD, 0);

    hipLaunchKernelGGL(hgt_init_layer, dim3(gNH), blk, 0, stream, aggr, zsum, mEnc, N);

    hipLaunchKernelGGL(hgt_edge_score, gEdge, blk, 0, stream,
                       esrc, edst, edge_type, edge_time, node_type,
                       q, k, k_rte, ratt_l, rpri_l, score, mEnc, E);
    hipLaunchKernelGGL(hgt_edge_expsum, dim3(gEH), blk, 0, stream,
                       edst, score, mEnc, zsum, E);
    hipLaunchKernelGGL(hgt_edge_msg, gEdge, blk, 0, stream,
                       esrc, edst, edge_type, edge_time, node_type,
                       v, v_rte, rmsg_l, score, mEnc, zsum, aggr, E);

    hipLaunchKernelGGL(hgt_gelu, dim3(gNH), blk, 0, stream, aggr, (size_t)N * HD);

    hipLaunchKernelGGL(hgt_wmma_linear, gN32T, blk, 0, stream,
                       aggr, Wa_l, ba_l, node_type, outb, N, HD, 0);

    hipLaunchKernelGGL(hgt_blend, dim3(gNH), blk, 0, stream,
                       outb, node_type, skip + (size_t)l * TYP, x, N);
  }
}